// ResContextBlock_63591285785183
// MI455X (gfx1250) — compile-verified
//
#include <hip/hip_runtime.h>
#include <hip/hip_bf16.h>

// ---------------------------------------------------------------------------
// ResContextBlock on gfx1250 (MI455X): 4 gather-GEMMs via V_WMMA_F32_16X16X4_F32.
// - BN folded into the consuming conv (scale into LDS-staged weights, bias via
//   3 extra WMMA K-steps driven by a tap-validity bitmask).
// - 32-point M-tile per wave: every B fragment feeds 4 WMMAs, 4 independent
//   accumulator chains keep the matrix pipe busy under gather latency.
// - Tap loop force-unrolled with an index-prefetch pass so each wave keeps
//   dozens of random gathers in flight (MLP >> rolled-loop serialization).
// - Invalid neighbors gather from a zeroed row in workspace (no per-element
//   cndmasks), matching where(valid, x, 0) exactly.
// ---------------------------------------------------------------------------

typedef __attribute__((ext_vector_type(2))) float v2f;
typedef __attribute__((ext_vector_type(4))) float v4f;
typedef __attribute__((ext_vector_type(8))) float v8f;

#define NPTS   250000
#define COUT   32
#define NTAPS  9
#define SLOPE  0.01f
#define BN_EPS 1e-3f

__device__ __forceinline__ v8f wmma_f32(v2f a, v2f b, v8f c) {
    return __builtin_amdgcn_wmma_f32_16x16x4_f32(
        false, a, false, b, (short)0, c, false, false);
}

// A (16x4 f32, 2 VGPR): lane m=lane&15 is row m; lanes 0-15 hold local K {0,1},
// lanes 16-31 hold local K {2,3}.  B mirrors A with lane = column.
// K-row permutation: step s = off*(CIN/4) + 2g + t covers cin = 8g+4*hi+2t+{0,1},
// so one global_load_b128 per lane feeds two K-steps with no wasted bytes.
template <int CIN, bool AFF>
__global__ __launch_bounds__(256) void subm_conv_wmma(
    const float* __restrict__ src,   // [NPTS, CIN]
    const int*   __restrict__ nbr,   // [NTAPS, NPTS], -1 = no neighbor
    const float* __restrict__ w,     // [NTAPS, CIN, COUT]
    const float* __restrict__ aff,   // [2][COUT]: scale, bias (unused if !AFF)
    const float* __restrict__ zrow,  // [COUT] zeros: gather target for invalid taps
    float*       __restrict__ dst,   // [NPTS, COUT] lrelu(conv) output
    float*       __restrict__ stat,  // [2][COUT] global sum/sumsq accumulators
    int n)
{
    constexpr int KMAIN = NTAPS * CIN;          // 144 / 288
    constexpr int SMAIN = KMAIN / 4;            // 36 / 72 main K-steps
    constexpr int STEPS = SMAIN + (AFF ? 3 : 0);

    __shared__ float pk[STEPS * 128];           // [step][lane][4] packed B
    __shared__ float s_sum[COUT];
    __shared__ float s_ssq[COUT];

    const int tid = threadIdx.x;

    // ---- Stage B into LDS: scale-folded weights + per-tap bias rows --------
    for (int i = tid; i < STEPS * 128; i += 256) {
        const int s   = i >> 7;
        const int l   = (i >> 2) & 31;
        const int j   = i & 3;
        const int hi_ = l >> 4;
        const int col = (l & 15) + ((j >> 1) << 4);
        float val;
        if (s < SMAIN) {
            const int off = s / (CIN / 4);
            const int r   = s % (CIN / 4);
            const int cin = 8 * (r >> 1) + 4 * hi_ + 2 * (r & 1) + (j & 1);
            val = w[(off * CIN + cin) * COUT + col];
            if (AFF) val *= aff[cin];
        } else {
            const int off_b = (s - SMAIN) * 4 + 2 * hi_ + (j & 1);
            val = 0.0f;
            if (AFF && off_b < NTAPS) {         // bias row: sum_c c[cin]*W[tap][cin][col]
                float acc = 0.0f;
                for (int cin = 0; cin < CIN; ++cin)
                    acc += aff[COUT + cin] * w[(off_b * CIN + cin) * COUT + col];
                val = acc;
            }
        }
        pk[i] = val;
    }
    if (tid < COUT) { s_sum[tid] = 0.0f; s_ssq[tid] = 0.0f; }
    __syncthreads();

    const int wave = tid >> 5;
    const int lane = tid & 31;
    const int p0   = (blockIdx.x * 8 + wave) * 32;   // 32-point super-tile
    const bool act0 = (p0 < n);
    const bool act1 = (p0 + 16 < n);

    const int m  = lane & 15;
    const int hi = lane >> 4;

    // ---- Phase 1: prefetch all neighbor indices (maximal load clause) ------
    int idx0[NTAPS], idx1[NTAPS];
#pragma unroll
    for (int off = 0; off < NTAPS; ++off) {
        idx0[off] = act0 ? nbr[off * n + p0 + m]      : -1;
        idx1[off] = act1 ? nbr[off * n + p0 + 16 + m] : -1;
    }
    unsigned vmask0 = 0, vmask1 = 0;
#pragma unroll
    for (int off = 0; off < NTAPS; ++off) {
        vmask0 |= (idx0[off] >= 0 ? 1u : 0u) << off;
        vmask1 |= (idx1[off] >= 0 ? 1u : 0u) << off;
    }

    v8f acc00 = {};   // point-tile 0, output cols 0..15
    v8f acc01 = {};   // point-tile 0, output cols 16..31
    v8f acc10 = {};   // point-tile 1, output cols 0..15
    v8f acc11 = {};   // point-tile 1, output cols 16..31

    // ---- Phase 2: fully unrolled gather + WMMA (deep MLP per wave) ---------
#pragma unroll
    for (int off = 0; off < NTAPS; ++off) {
        // invalid taps gather a zeroed row (WGP$-resident) -> no per-elem selects
        const float* rowp0 =
            ((idx0[off] >= 0) ? (src + (long)idx0[off] * CIN) : zrow) + 4 * hi;
        const float* rowp1 =
            ((idx1[off] >= 0) ? (src + (long)idx1[off] * CIN) : zrow) + 4 * hi;

#pragma unroll
        for (int g = 0; g < CIN / 8; ++g) {
            const v4f gx0 = *(const v4f*)(rowp0 + 8 * g);
            const v4f gx1 = *(const v4f*)(rowp1 + 8 * g);

            const int sbase = (off * (CIN / 4) + 2 * g) * 128 + lane * 4;
            const v4f bA = *(const v4f*)(&pk[sbase]);        // step 2g  : cols {m,16+m}
            const v4f bB = *(const v4f*)(&pk[sbase + 128]);  // step 2g+1: cols {m,16+m}

            v2f a00; a00.x = gx0.x; a00.y = gx0.y;
            v2f a01; a01.x = gx0.z; a01.y = gx0.w;
            v2f a10; a10.x = gx1.x; a10.y = gx1.y;
            v2f a11; a11.x = gx1.z; a11.y = gx1.w;
            v2f b00; b00.x = bA.x; b00.y = bA.y;
            v2f b01; b01.x = bA.z; b01.y = bA.w;
            v2f b10; b10.x = bB.x; b10.y = bB.y;
            v2f b11; b11.x = bB.z; b11.y = bB.w;

            // 4 independent accumulator chains, B reused 4x per ds_load_b128
            acc00 = wmma_f32(a00, b00, acc00);
            acc01 = wmma_f32(a00, b01, acc01);
            acc10 = wmma_f32(a10, b00, acc10);
            acc11 = wmma_f32(a10, b01, acc11);
            acc00 = wmma_f32(a01, b10, acc00);
            acc01 = wmma_f32(a01, b11, acc01);
            acc10 = wmma_f32(a11, b10, acc10);
            acc11 = wmma_f32(a11, b11, acc11);
        }
    }

    if (AFF) {
        // Bias contribution: [16x12 validity 0/1] x [12x32 bias rows], per tile
#pragma unroll
        for (int s = 0; s < 3; ++s) {
            const int rb = 4 * s + 2 * hi;
            v2f a0, a1;
            a0.x = (float)((vmask0 >> rb) & 1u);
            a0.y = (float)((vmask0 >> (rb + 1)) & 1u);
            a1.x = (float)((vmask1 >> rb) & 1u);
            a1.y = (float)((vmask1 >> (rb + 1)) & 1u);
            const int sbase = (SMAIN + s) * 128 + lane * 4;
            const v4f bA = *(const v4f*)(&pk[sbase]);
            v2f b0; b0.x = bA.x; b0.y = bA.y;
            v2f b1; b1.x = bA.z; b1.y = bA.w;
            acc00 = wmma_f32(a0, b0, acc00);
            acc01 = wmma_f32(a0, b1, acc01);
            acc10 = wmma_f32(a1, b0, acc10);
            acc11 = wmma_f32(a1, b1, acc11);
        }
    }

    // ---- LeakyReLU, store, per-channel stats for the consuming BN ----------
    // C/D layout: VGPR r holds M = r + 8*hi, column m (coltile0) / 16+m (coltile1).
    float s0 = 0.f, q0 = 0.f, s1 = 0.f, q1 = 0.f;
    if (act0) {
#pragma unroll
        for (int r = 0; r < 8; ++r) {
            const int row = p0 + r + hi * 8;
            float y0 = acc00[r];
            float y1 = acc01[r];
            y0 = (y0 >= 0.f) ? y0 : SLOPE * y0;
            y1 = (y1 >= 0.f) ? y1 : SLOPE * y1;
            dst[row * COUT + m]      = y0;
            dst[row * COUT + 16 + m] = y1;
            s0 += y0; q0 += y0 * y0;
            s1 += y1; q1 += y1 * y1;
        }
    }
    if (act1) {
#pragma unroll
        for (int r = 0; r < 8; ++r) {
            const int row = p0 + 16 + r + hi * 8;
            float y0 = acc10[r];
            float y1 = acc11[r];
            y0 = (y0 >= 0.f) ? y0 : SLOPE * y0;
            y1 = (y1 >= 0.f) ? y1 : SLOPE * y1;
            dst[row * COUT + m]      = y0;
            dst[row * COUT + 16 + m] = y1;
            s0 += y0; q0 += y0 * y0;
            s1 += y1; q1 += y1 * y1;
        }
    }
    if (act0) {
        atomicAdd(&s_sum[m], s0);
        atomicAdd(&s_ssq[m], q0);
        atomicAdd(&s_sum[16 + m], s1);
        atomicAdd(&s_ssq[16 + m], q1);
    }
    __syncthreads();

    if (tid < COUT) {
        atomicAdd(&stat[tid], s_sum[tid]);
        atomicAdd(&stat[COUT + tid], s_ssq[tid]);
    }
}

// stats -> folded BN affine: a = g * rsqrt(var+eps), c = b - mean * a
__global__ void bn_affine_kernel(const float* __restrict__ stat,
                                 const float* __restrict__ g,
                                 const float* __restrict__ b,
                                 float* __restrict__ aff,
                                 float invN)
{
    const int c = threadIdx.x;
    if (c < COUT) {
        const float mean = stat[c] * invN;
        const float var  = stat[COUT + c] * invN - mean * mean;
        const float a    = g[c] * rsqrtf(var + BN_EPS);
        aff[c]        = a;
        aff[COUT + c] = b[c] - mean * a;
    }
}

// out = bn2(t3) + bn02(t1), with t3 staged in d_out (in-place affine).
__global__ void final_add_kernel(float* __restrict__ out,
                                 const float* __restrict__ t1,
                                 const float* __restrict__ aff2,
                                 const float* __restrict__ aff02,
                                 int total)
{
    const int i = blockIdx.x * blockDim.x + threadIdx.x;
    if (i < total) {
        const int c = i & (COUT - 1);
        out[i] = aff2[c] * out[i] + aff2[COUT + c]
               + aff02[c] * t1[i] + aff02[COUT + c];
    }
}

extern "C" void kernel_launch(void* const* d_in, const int* in_sizes, int n_in,
                              void* d_out, int out_size, void* d_ws, size_t ws_size,
                              hipStream_t stream)
{
    (void)in_sizes; (void)n_in; (void)out_size; (void)ws_size;

    const float* features = (const float*)d_in[0];
    const int*   nbr133   = (const int*)d_in[1];
    const int*   nbr313   = (const int*)d_in[2];
    const float* conv1_w  = (const float*)d_in[3];
    const float* conv12_w = (const float*)d_in[4];
    const float* conv2_w  = (const float*)d_in[5];
    const float* conv3_w  = (const float*)d_in[6];
    const float* bn0_g  = (const float*)d_in[7],  *bn0_b  = (const float*)d_in[8];
    const float* bn02_g = (const float*)d_in[9],  *bn02_b = (const float*)d_in[10];
    const float* bn1_g  = (const float*)d_in[11], *bn1_b  = (const float*)d_in[12];
    const float* bn2_g  = (const float*)d_in[13], *bn2_b  = (const float*)d_in[14];

    float* out = (float*)d_out;

    const size_t N32 = (size_t)NPTS * COUT;
    float* tA = (float*)d_ws;          // t0, later reused as t2
    float* tB = tA + N32;              // t1 (live until the end)
    float* stat0  = tB + N32;          // 4 x [2][32] stats
    float* stat02 = stat0  + 2 * COUT;
    float* stat1  = stat02 + 2 * COUT;
    float* stat2  = stat1  + 2 * COUT;
    float* zrow   = stat2  + 2 * COUT; // [32] zeros: invalid-tap gather target
    float* aff0   = zrow   + COUT;     // 4 x [2][32] folded-BN affines
    float* aff02  = aff0   + 2 * COUT;
    float* aff1   = aff02  + 2 * COUT;
    float* aff2   = aff1   + 2 * COUT;

    // zero stats (4x64) + zero row (32) in one memset node
    hipMemsetAsync(stat0, 0, (8 * COUT + COUT) * sizeof(float), stream);

    const int blocks = (NPTS + 255) / 256;        // 8 waves x 32 points per block
    const float invN = 1.0f / (float)NPTS;
    const int total  = (int)N32;

    // shortcut branch: conv1(1x3x3) -> lrelu -> bn0 -> conv12(3x1x3) -> lrelu -> bn02
    subm_conv_wmma<16, false><<<blocks, 256, 0, stream>>>(features, nbr133, conv1_w,
                                                          nullptr, zrow, tA, stat0, NPTS);
    bn_affine_kernel<<<1, COUT, 0, stream>>>(stat0, bn0_g, bn0_b, aff0, invN);
    subm_conv_wmma<32, true><<<blocks, 256, 0, stream>>>(tA, nbr313, conv12_w,
                                                         aff0, zrow, tB, stat02, NPTS);
    bn_affine_kernel<<<1, COUT, 0, stream>>>(stat02, bn02_g, bn02_b, aff02, invN);

    // main branch: conv2(3x1x3) -> lrelu -> bn1 -> conv3(1x3x3) -> lrelu -> bn2
    subm_conv_wmma<16, false><<<blocks, 256, 0, stream>>>(features, nbr313, conv2_w,
                                                          nullptr, zrow, tA, stat1, NPTS);
    bn_affine_kernel<<<1, COUT, 0, stream>>>(stat1, bn1_g, bn1_b, aff1, invN);
    subm_conv_wmma<32, true><<<blocks, 256, 0, stream>>>(tA, nbr133, conv3_w,
                                                         aff1, zrow, out, stat2, NPTS);
    bn_affine_kernel<<<1, COUT, 0, stream>>>(stat2, bn2_g, bn2_b, aff2, invN);

    // out = bn2(t3) + bn02(t1)
    final_add_kernel<<<(total + 255) / 256, 256, 0, stream>>>(out, tB, aff2, aff02, total);
}